// RootModel_28913719837232
// MI455X (gfx1250) — compile-verified
//
#include <hip/hip_runtime.h>

#define NN 50000
#define NE 800000
#define DIM 128

#define AS1 __attribute__((address_space(1)))

typedef float v2f __attribute__((ext_vector_type(2)));
typedef float v4f __attribute__((ext_vector_type(4)));
typedef float v8f __attribute__((ext_vector_type(8)));

typedef const AS1 float* gcf;
typedef AS1 float*       gf;
typedef const AS1 v2f*   gc2;
typedef const AS1 v4f*   gc4;

// ---------------------------------------------------------------------------
// Fused node GEMM: K = X@Wk^T+bk, Q = X@Wq^T+bq, V = X@Wv^T+bv,
//                  skip = X@Ws^T+bconv  (skip written to layer output buffer;
//                  edge kernel atomically accumulates messages on top of it)
// One block = 16 rows of X staged in LDS; 8 waves x 4 tiles = 32 (mat,colblk)
// 16x16 output tiles, each via 32x V_WMMA_F32_16X16X4_F32 (full fp32).
// All global accesses forced to addrspace(1) -> global_load/store, not flat.
// ---------------------------------------------------------------------------
__global__ __launch_bounds__(256) void gemm_kqvs(
    const float* __restrict__ X,
    const float* __restrict__ Wk, const float* __restrict__ bk,
    const float* __restrict__ Wq, const float* __restrict__ bq,
    const float* __restrict__ Wv, const float* __restrict__ bv,
    const float* __restrict__ Ws, const float* __restrict__ bs,
    float* __restrict__ Ko, float* __restrict__ Qo,
    float* __restrict__ Vo, float* __restrict__ So)
{
    __shared__ float sX[16 * 132];           // pad 128->132: conflict-free WMMA reads
    const int tid = threadIdx.x;
    const int r0  = blockIdx.x * 16;         // 50000 = 3125 * 16 exactly

    // Stage 16x128 X tile: 512 v4f total, 2 per thread, b128 global + b128 ds.
    for (int i = 0; i < 2; ++i) {
        const int idx = tid + 256 * i;       // 0..511
        const int row = idx >> 5;            // 32 v4f per row
        const int c4  = idx & 31;
        const v4f x4 = ((gc4)(X + (size_t)(r0 + row) * DIM))[c4];
        *(v4f*)&sX[row * 132 + c4 * 4] = x4;
    }
    __syncthreads();

    const int wave = tid >> 5;
    const int lane = tid & 31;
    const int half = lane >> 4;              // 0: lanes 0-15, 1: lanes 16-31
    const int lan  = lane & 15;

    const float* Wm[4] = {Wk, Wq, Wv, Ws};
    const float* Bm[4] = {bk, bq, bv, bs};
    float*       Om[4] = {Ko, Qo, Vo, So};

    for (int tt = 0; tt < 4; ++tt) {
        const int t   = wave * 4 + tt;       // 0..31
        const int mat = t >> 3;              // which of K/Q/V/skip
        const int n0  = (t & 7) * 16;        // output column block
        gcf W = (gcf)Wm[mat];

        v8f c = {0.f, 0.f, 0.f, 0.f, 0.f, 0.f, 0.f, 0.f};
        for (int kb = 0; kb < 32; ++kb) {
            const int k = kb * 4 + 2 * half;
            // A (16x4 f32): lane holds {X[m][k], X[m][k+1]}, m = lane&15
            v2f a;
            a.x = sX[lan * 132 + k];
            a.y = sX[lan * 132 + k + 1];
            // B (4x16 f32): B[kk][n] = W[n][kk]  (Y = X @ W^T), 8B-aligned
            const v2f b = *(gc2)(W + (size_t)(n0 + lan) * DIM + k);
            c = __builtin_amdgcn_wmma_f32_16x16x4_f32(
                    false, a, false, b, (short)0, c, false, false);
        }
        // C/D layout: VGPR r -> (M = r + 8*half, N = lane&15)
        const float bias = ((gcf)Bm[mat])[n0 + lan];
        gf O = (gf)Om[mat];
        for (int r = 0; r < 8; ++r) {
            const int row = r0 + r + 8 * half;
            O[(size_t)row * DIM + n0 + lan] = c[r] + bias;
        }
    }
}

// ---------------------------------------------------------------------------
// Edge stage: one wave32 per edge; lane owns a v4f feature slice.
// gate = sigmoid(K[dst] + Q[src]); Out[dst] += gate * V[src]  (f32 atomics;
// K/Q/V rows are L2-resident on MI455X: 77 MB << 192 MB L2).
// ---------------------------------------------------------------------------
__global__ __launch_bounds__(256) void edge_msg(
    const long long* __restrict__ ei,
    const float* __restrict__ Kb, const float* __restrict__ Qb,
    const float* __restrict__ Vb, float* __restrict__ Out)
{
    const int lane = threadIdx.x & 31;
    long long w  = (long long)blockIdx.x * (blockDim.x >> 5) + (threadIdx.x >> 5);
    const long long nw = (long long)gridDim.x * (blockDim.x >> 5);

    for (long long e = w; e < NE; e += nw) {
        const long long s = ((const AS1 long long*)ei)[e];
        const long long d = ((const AS1 long long*)ei)[NE + e];
        const v4f q4 = ((gc4)(Qb + s * DIM))[lane];
        const v4f k4 = ((gc4)(Kb + d * DIM))[lane];
        const v4f v4 = ((gc4)(Vb + s * DIM))[lane];
        float* o = Out + d * DIM + lane * 4;
        const float gx = 1.f / (1.f + __expf(-(k4.x + q4.x)));
        const float gy = 1.f / (1.f + __expf(-(k4.y + q4.y)));
        const float gz = 1.f / (1.f + __expf(-(k4.z + q4.z)));
        const float gw = 1.f / (1.f + __expf(-(k4.w + q4.w)));
        atomicAdd(o + 0, gx * v4.x);
        atomicAdd(o + 1, gy * v4.y);
        atomicAdd(o + 2, gz * v4.z);
        atomicAdd(o + 3, gw * v4.w);
    }
}

// --------------------------- Set2Set kernels -------------------------------
__global__ __launch_bounds__(256) void dot_max(
    const float* __restrict__ X, const float* __restrict__ h,
    float* __restrict__ e, unsigned* __restrict__ emax)
{
    const int lane = threadIdx.x & 31;
    int w  = blockIdx.x * (blockDim.x >> 5) + (threadIdx.x >> 5);
    int nw = gridDim.x * (blockDim.x >> 5);
    const v4f h4 = ((gc4)h)[lane];
    for (int i = w; i < NN; i += nw) {
        const v4f x4 = ((gc4)(X + (size_t)i * DIM))[lane];
        float s = x4.x * h4.x + x4.y * h4.y + x4.z * h4.z + x4.w * h4.w;
        for (int o = 16; o > 0; o >>= 1) s += __shfl_xor(s, o);
        if (lane == 0) {
            e[i] = s;
            unsigned u = __float_as_uint(s);
            u = (u & 0x80000000u) ? ~u : (u | 0x80000000u);  // order-preserving
            atomicMax(emax, u);
        }
    }
}

__global__ __launch_bounds__(256) void softmax_accum(
    const float* __restrict__ X, const float* __restrict__ e,
    const unsigned* __restrict__ emaxU,
    float* __restrict__ R, float* __restrict__ S)
{
    const int lane = threadIdx.x & 31;
    int w  = blockIdx.x * (blockDim.x >> 5) + (threadIdx.x >> 5);
    int nw = gridDim.x * (blockDim.x >> 5);
    const unsigned u = *emaxU;
    const float m = (u & 0x80000000u) ? __uint_as_float(u ^ 0x80000000u)
                                      : __uint_as_float(~u);
    v4f acc = {0.f, 0.f, 0.f, 0.f};
    float ps = 0.f;
    for (int i = w; i < NN; i += nw) {
        const float p = __expf(((gcf)e)[i] - m);
        const v4f x4 = ((gc4)(X + (size_t)i * DIM))[lane];
        acc.x += p * x4.x; acc.y += p * x4.y;
        acc.z += p * x4.z; acc.w += p * x4.w;
        if (lane == 0) ps += p;
    }
    atomicAdd(&R[lane * 4 + 0], acc.x);
    atomicAdd(&R[lane * 4 + 1], acc.y);
    atomicAdd(&R[lane * 4 + 2], acc.z);
    atomicAdd(&R[lane * 4 + 3], acc.w);
    if (lane == 0) atomicAdd(S, ps);
}

__global__ __launch_bounds__(512) void lstm_step(
    const float* __restrict__ Wih, const float* __restrict__ bih,
    const float* __restrict__ Whh, const float* __restrict__ bhh,
    float* __restrict__ h, float* __restrict__ c,
    const float* __restrict__ qstar)
{
    __shared__ float gates[512];
    const int g = threadIdx.x;
    float acc = bih[g] + bhh[g];
    const float* wi = Wih + (size_t)g * 256;
    for (int j = 0; j < 256; ++j) acc += wi[j] * qstar[j];
    const float* wh = Whh + (size_t)g * 128;
    for (int j = 0; j < 128; ++j) acc += wh[j] * h[j];
    gates[g] = acc;
    __syncthreads();
    if (g < 128) {
        const float ig = gates[g], fg = gates[128 + g];
        const float gg = gates[256 + g], og = gates[384 + g];
        const float si = 1.f / (1.f + __expf(-ig));
        const float sf = 1.f / (1.f + __expf(-fg));
        const float so = 1.f / (1.f + __expf(-og));
        const float cn = sf * c[g] + si * tanhf(gg);
        c[g] = cn;
        h[g] = so * tanhf(cn);
    }
}

__global__ void build_qstar(const float* __restrict__ h,
                            const float* __restrict__ R,
                            const float* __restrict__ S,
                            float* __restrict__ qstar)
{
    const int t = threadIdx.x;        // 128 threads
    const float s = *S;
    qstar[t] = h[t];
    qstar[128 + t] = R[t] / s;
}

__global__ void final_linear(const float* __restrict__ Wo,
                             const float* __restrict__ bo,
                             const float* __restrict__ qstar,
                             float* __restrict__ out)
{
    const int d = threadIdx.x;        // 128 threads
    float acc = bo[d];
    const float* w = Wo + (size_t)d * 256;
    for (int j = 0; j < 256; ++j) acc += w[j] * qstar[j];
    out[d] = acc;
}

__global__ void zero_state(float* __restrict__ h, float* __restrict__ c,
                           float* __restrict__ qstar)
{
    const int t = threadIdx.x;        // 512 threads
    if (t < 128)      h[t] = 0.f;
    else if (t < 256) c[t - 128] = 0.f;
    else              qstar[t - 256] = 0.f;
}

__global__ void zero_step(float* __restrict__ R, float* __restrict__ S,
                          unsigned* __restrict__ emax)
{
    const int t = threadIdx.x;        // 256 threads
    if (t < 128) R[t] = 0.f;
    if (t == 128) *S = 0.f;
    if (t == 129) *emax = 0u;
}

// ---------------------------------------------------------------------------
extern "C" void kernel_launch(void* const* d_in, const int* in_sizes, int n_in,
                              void* d_out, int out_size, void* d_ws, size_t ws_size,
                              hipStream_t stream)
{
    const float*     x   = (const float*)d_in[0];
    const long long* ei  = (const long long*)d_in[1];
    // d_in[2] = edge_attr: unused by the reference
    const float* Wk  = (const float*)d_in[3];
    const float* bk  = (const float*)d_in[4];
    const float* Wq  = (const float*)d_in[5];
    const float* bq  = (const float*)d_in[6];
    const float* Wv  = (const float*)d_in[7];
    const float* bv  = (const float*)d_in[8];
    const float* Ws  = (const float*)d_in[9];
    const float* bc  = (const float*)d_in[10];
    const float* Wih = (const float*)d_in[11];
    const float* bih = (const float*)d_in[12];
    const float* Whh = (const float*)d_in[13];
    const float* bhh = (const float*)d_in[14];
    const float* Wo  = (const float*)d_in[15];
    const float* bo  = (const float*)d_in[16];
    float* out = (float*)d_out;

    const size_t nd = (size_t)NN * DIM;
    float* ws    = (float*)d_ws;
    float* x1    = ws;            // layer-1 output (skip + agg)
    float* x2    = x1 + nd;       // layer-2 output
    float* Kb    = x2 + nd;
    float* Qb    = Kb + nd;
    float* Vb    = Qb + nd;
    float* ebuf  = Vb + nd;       // attention logits [N]
    float* R     = ebuf + NN;     // 128
    float* S     = R + 128;       // 1
    unsigned* em = (unsigned*)(S + 1);
    float* h     = (float*)(em + 1);
    float* c     = h + 128;
    float* qstar = c + 128;       // 256

    const dim3 blk(256);
    const size_t wstep = (size_t)DIM * DIM;

    // ---- layer 0 ----
    gemm_kqvs<<<NN / 16, blk, 0, stream>>>(x, Wk, bk, Wq, bq, Wv, bv, Ws, bc,
                                           Kb, Qb, Vb, x1);
    edge_msg<<<4800, blk, 0, stream>>>(ei, Kb, Qb, Vb, x1);
    // ---- layer 1 ----
    gemm_kqvs<<<NN / 16, blk, 0, stream>>>(x1, Wk + wstep, bk + DIM,
                                           Wq + wstep, bq + DIM,
                                           Wv + wstep, bv + DIM,
                                           Ws + wstep, bc + DIM,
                                           Kb, Qb, Vb, x2);
    edge_msg<<<4800, blk, 0, stream>>>(ei, Kb, Qb, Vb, x2);

    // ---- Set2Set (batch 1, 3 steps) ----
    zero_state<<<1, 512, 0, stream>>>(h, c, qstar);
    for (int s3 = 0; s3 < 3; ++s3) {
        lstm_step<<<1, 512, 0, stream>>>(Wih, bih, Whh, bhh, h, c, qstar);
        zero_step<<<1, 256, 0, stream>>>(R, S, em);
        dot_max<<<6250, blk, 0, stream>>>(x2, h, ebuf, em);
        softmax_accum<<<1024, blk, 0, stream>>>(x2, ebuf, em, R, S);
        build_qstar<<<1, 128, 0, stream>>>(h, R, S, qstar);
    }
    final_linear<<<1, 128, 0, stream>>>(Wo, bo, qstar, out);
}